// DRAMAttention_8675833938083
// MI455X (gfx1250) — compile-verified
//
#include <hip/hip_runtime.h>

typedef __attribute__((ext_vector_type(2))) float v2f;
typedef __attribute__((ext_vector_type(8))) float v8f;

#define S_Q    32
#define HQ     32
#define NKV    8
#define DHEAD  128
#define GROUP  4
#define PAGESZ 64
#define NPAGES 1024
#define TOPK   64
#define LLOC   4096
#define NWAVE  4                       /* split-K waves per block */
#define QKSCALE 0.08838834764831845f   /* 1/sqrt(128) */

// LDS float-index layout (per block)
#define STRIDEK 132                    /* padded row stride (floats) */
#define MATF    (16 * STRIDEK)         /* 2112 floats per staged matrix */
#define WSTAGE  (4 * MATF)             /* per-wave staging: 2 bufs x (K,V) */
#define OFF_SP  (NWAVE * WSTAGE)       /* 33792 : P relayout, 288/wave */
#define OFF_ML  (OFF_SP + NWAVE * 288) /* 34944 : (m,l) per wave/row      */
#define OFF_SC  (OFF_ML + NWAVE * 32)  /* 35072 : merge scales            */
#define OFF_LS  (OFF_SC + NWAVE * 16)  /* 35136 : merged l per row        */
#define OFF_LE  (OFF_LS + 16)          /* 35152 : merged lse per row      */
#define SMEMF   (OFF_LE + 16)          /* 35168 floats = 140672 B         */

// ---------------------------------------------------------------------------
// qsum[kvh][d] = sum_{s,g} xq[s][kvh*4+g][d]
// ---------------------------------------------------------------------------
__global__ __launch_bounds__(256) void qsum_kernel(const float* __restrict__ xq,
                                                   float* __restrict__ qsum) {
  for (int o = threadIdx.x; o < NKV * DHEAD; o += 256) {
    int kvh = o / DHEAD, d = o % DHEAD;
    float acc = 0.f;
    for (int s = 0; s < S_Q; ++s)
      for (int g = 0; g < GROUP; ++g)
        acc += xq[((s * HQ) + (kvh * GROUP + g)) * DHEAD + d];
    qsum[o] = acc;
  }
}

// ---------------------------------------------------------------------------
// scores[p] = (1/64) * <page elements, qsum broadcast>  -- 256 MB b128 stream
// ---------------------------------------------------------------------------
__global__ __launch_bounds__(256) void score_kernel(const float* __restrict__ dram_k,
                                                    const float* __restrict__ qsum,
                                                    float* __restrict__ scores) {
  __shared__ float qs[NKV * DHEAD];
  __shared__ float red[256];
  int p = blockIdx.x;
  for (int i = threadIdx.x; i < NKV * DHEAD; i += 256) qs[i] = qsum[i];
  __syncthreads();
  // page data is 65536 contiguous floats; element e pairs with qs[e & 1023].
  // For float4 index f = tid + it*256: (4f) & 1023 == 4*tid  (invariant).
  const float4* base4 = (const float4*)(dram_k + (size_t)p * (PAGESZ * NKV * DHEAD));
  float4 qv = *(const float4*)(&qs[4 * threadIdx.x]);
  float acc = 0.f;
  for (int it = 0; it < 64; ++it) {
    float4 kv = base4[threadIdx.x + it * 256];
    acc += kv.x * qv.x + kv.y * qv.y + kv.z * qv.z + kv.w * qv.w;
  }
  red[threadIdx.x] = acc;
  __syncthreads();
  for (int off = 128; off > 0; off >>= 1) {
    if (threadIdx.x < off) red[threadIdx.x] += red[threadIdx.x + off];
    __syncthreads();
  }
  if (threadIdx.x == 0) scores[p] = red[0] * (1.0f / PAGESZ);
}

// ---------------------------------------------------------------------------
// Top-64 of 1024 scores (ties -> lower index, like jax.lax.top_k), sorted asc.
// ---------------------------------------------------------------------------
__global__ __launch_bounds__(256) void topk_kernel(const float* __restrict__ scores,
                                                   int* __restrict__ idx_out) {
  __shared__ float sv[NPAGES];
  __shared__ float rv[256];
  __shared__ int   ri[256];
  __shared__ int   chosen[TOPK];
  const float NEG_INF = -__builtin_huge_valf();
  for (int i = threadIdx.x; i < NPAGES; i += 256) sv[i] = scores[i];
  __syncthreads();
  for (int it = 0; it < TOPK; ++it) {
    float bv = NEG_INF; int bi = NPAGES;
    for (int i = threadIdx.x; i < NPAGES; i += 256) {
      float v = sv[i];
      if (v > bv || (v == bv && i < bi)) { bv = v; bi = i; }
    }
    rv[threadIdx.x] = bv; ri[threadIdx.x] = bi;
    __syncthreads();
    for (int off = 128; off > 0; off >>= 1) {
      if (threadIdx.x < off) {
        float v2 = rv[threadIdx.x + off]; int i2 = ri[threadIdx.x + off];
        if (v2 > rv[threadIdx.x] || (v2 == rv[threadIdx.x] && i2 < ri[threadIdx.x])) {
          rv[threadIdx.x] = v2; ri[threadIdx.x] = i2;
        }
      }
      __syncthreads();
    }
    if (threadIdx.x == 0) { chosen[it] = ri[0]; sv[ri[0]] = NEG_INF; }
    __syncthreads();
  }
  if (threadIdx.x == 0) {
    for (int i = 1; i < TOPK; ++i) {
      int key = chosen[i]; int j = i - 1;
      while (j >= 0 && chosen[j] > key) { chosen[j + 1] = chosen[j]; --j; }
      chosen[j + 1] = key;
    }
    for (int i = 0; i < TOPK; ++i) idx_out[i] = chosen[i];
  }
}

// ---------------------------------------------------------------------------
// Issue one 16-key chunk of K and V into LDS via GLOBAL_LOAD_ASYNC_TO_LDS_B128
// (GVS mode: uniform SGPR row base + lane*16B offset). 32 async ops / chunk.
// ---------------------------------------------------------------------------
__device__ __forceinline__ void issue_chunk_async(
    int c, int mode, int kvh, int lane,
    const float* __restrict__ main_k, const float* __restrict__ main_v,
    const float* __restrict__ tail_k, const float* __restrict__ tail_v,
    const int* __restrict__ page_idx,
    unsigned ldsK, unsigned ldsV) {
  long tok0; const float *kb, *vb;
  if (mode == 0) {
    int page = page_idx[c >> 2];                 // 16-key chunk stays in 1 page
    tok0 = (long)page * PAGESZ + (long)(c & 3) * 16;
    kb = main_k; vb = main_v;
  } else {
    int j0 = c * 16;
    if (j0 < LLOC) { tok0 = j0;        kb = main_k; vb = main_v; }
    else           { tok0 = j0 - LLOC; kb = tail_k; vb = tail_v; }
  }
  const float* kr0 = kb + (tok0 * NKV + kvh) * DHEAD;
  const float* vr0 = vb + (tok0 * NKV + kvh) * DHEAD;
  unsigned voff = (unsigned)lane * 16u;          // byte offset within row
#pragma unroll
  for (int j = 0; j < 16; ++j) {
    unsigned lk = ldsK + (unsigned)(j * STRIDEK * 4) + (unsigned)lane * 16u;
    unsigned lv = ldsV + (unsigned)(j * STRIDEK * 4) + (unsigned)lane * 16u;
    const float* krj = kr0 + j * (NKV * DHEAD);
    const float* vrj = vr0 + j * (NKV * DHEAD);
    asm volatile("global_load_async_to_lds_b128 %0, %1, %2"
                 :: "v"(lk), "v"(voff), "s"(krj) : "memory");
    asm volatile("global_load_async_to_lds_b128 %0, %1, %2"
                 :: "v"(lv), "v"(voff), "s"(vrj) : "memory");
  }
}

// ---------------------------------------------------------------------------
// Split-K flash attention. One block per (kv head, 16-query-row tile);
// NWAVE waves each handle every NWAVE-th 16-key chunk with async-LDS double
// buffering, then an in-LDS softmax merge combines the partials.
// mode 0: prefix (gathered pages) -> normalized out to d_out, lse to ws.
// mode 1: suffix (local + causal new) -> log-sum-exp merge with prefix.
// ---------------------------------------------------------------------------
__global__ __launch_bounds__(32 * NWAVE) void attn_kernel(
    const float* __restrict__ xq,
    const float* __restrict__ main_k,
    const float* __restrict__ main_v,
    const float* __restrict__ tail_k,
    const float* __restrict__ tail_v,
    const int*   __restrict__ page_idx,
    float* __restrict__ out,            // d_out  [S][H][D]
    float* __restrict__ lse_p,          // ws lse [8*128]
    int mode) {
  __shared__ float smem[SMEMF];

  const float NEG_INF = -__builtin_huge_valf();
  const int tid  = threadIdx.x;
  const int lane = tid & 31;
  const int w    = tid >> 5;               // wave id, 0..NWAVE-1
  const int kvh  = blockIdx.x >> 3;
  const int rt   = blockIdx.x & 7;
  const int mrow = lane & 15;              // M (A) / N (B,C,D) index
  const int koff = (lane < 16) ? 0 : 2;    // K sub-offset per lane half

  const unsigned smem_base = (unsigned)(size_t)(void*)smem;

  // --- preload Q fragments (A layout), fold in softmax scale --------------
  v2f qf[32];
  {
    int r = rt * 16 + mrow;                // r = g*32 + s
    int g = r >> 5, s = r & 31;
    const float* qrow = xq + ((s * HQ) + (kvh * GROUP + g)) * DHEAD;
#pragma unroll
    for (int kc = 0; kc < 32; ++kc) {
      v2f a;
      a.x = qrow[kc * 4 + koff]     * QKSCALE;
      a.y = qrow[kc * 4 + koff + 1] * QKSCALE;
      qf[kc] = a;
    }
  }

  v8f zero8 = {};
  v8f acc[8];
#pragma unroll
  for (int t = 0; t < 8; ++t) acc[t] = zero8;
  float m[8], lsum[8];
#pragma unroll
  for (int i = 0; i < 8; ++i) { m[i] = NEG_INF; lsum[i] = 0.f; }

  const int nchunk = (mode == 0) ? (TOPK * PAGESZ / 16) : ((LLOC + S_Q) / 16);

  // --- double-buffered async pipeline -------------------------------------
  const unsigned stage0 = smem_base + (unsigned)(w * WSTAGE) * 4u;
  issue_chunk_async(w, mode, kvh, lane, main_k, main_v, tail_k, tail_v,
                    page_idx, stage0, stage0 + MATF * 4u);
  int buf = 0;
  for (int c = w; c < nchunk; c += NWAVE) {
    int nc = c + NWAVE;
    if (nc < nchunk) {
      unsigned sb = stage0 + (unsigned)((buf ^ 1) * 2 * MATF) * 4u;
      issue_chunk_async(nc, mode, kvh, lane, main_k, main_v, tail_k, tail_v,
                        page_idx, sb, sb + MATF * 4u);
      asm volatile("s_wait_asynccnt 0x20" ::: "memory");  // current chunk done
    } else {
      asm volatile("s_wait_asynccnt 0x0" ::: "memory");
    }
    const float* sK = &smem[w * WSTAGE + buf * 2 * MATF];
    const float* sV = sK + MATF;

    // ---- S = (Q*scale) K^T : 32 wmma over D=128 --------------------------
    v8f st = {};
#pragma unroll
    for (int kc = 0; kc < 32; ++kc) {
      v2f b = *(const v2f*)(&sK[mrow * STRIDEK + kc * 4 + koff]);
      st = __builtin_amdgcn_wmma_f32_16x16x4_f32(false, qf[kc], false, b,
                                                 (short)0, st, false, false);
    }

    // ---- causal mask (suffix, last two chunks only) ----------------------
    if (mode == 1 && c * 16 >= LLOC) {
      int key_off = c * 16 - LLOC + mrow;
#pragma unroll
      for (int i = 0; i < 8; ++i) {
        int r = rt * 16 + ((lane < 16) ? i : i + 8);
        int s = r & 31;
        if (key_off > s) st[i] = -1e30f;
      }
    }

    // ---- online softmax --------------------------------------------------
    float alpha[8];
#pragma unroll
    for (int i = 0; i < 8; ++i) {
      float v = st[i];
      v = fmaxf(v, __shfl_xor(v, 1, 32));
      v = fmaxf(v, __shfl_xor(v, 2, 32));
      v = fmaxf(v, __shfl_xor(v, 4, 32));
      v = fmaxf(v, __shfl_xor(v, 8, 32));
      float mn = fmaxf(m[i], v);
      alpha[i] = __expf(m[i] - mn);
      m[i] = mn;
    }
#pragma unroll
    for (int i = 0; i < 8; ++i) {
      float p = __expf(st[i] - m[i]);
      st[i] = p;
      float rs = p;
      rs += __shfl_xor(rs, 1, 32);
      rs += __shfl_xor(rs, 2, 32);
      rs += __shfl_xor(rs, 4, 32);
      rs += __shfl_xor(rs, 8, 32);
      lsum[i] = lsum[i] * alpha[i] + rs;
    }
#pragma unroll
    for (int t = 0; t < 8; ++t)
#pragma unroll
      for (int i = 0; i < 8; ++i) acc[t][i] *= alpha[i];

    // ---- relayout P (C/D layout -> A layout) via per-wave LDS ------------
    float* sP = &smem[OFF_SP + w * 288];
#pragma unroll
    for (int i = 0; i < 8; ++i) {
      int r = (lane < 16) ? i : i + 8;
      sP[r * 18 + mrow] = st[i];
    }
    asm volatile("" ::: "memory");   // keep DS store->load order (HW in-order)
    v2f pf[4];
#pragma unroll
    for (int kc = 0; kc < 4; ++kc)
      pf[kc] = *(const v2f*)(&sP[mrow * 18 + kc * 4 + koff]);

    // ---- O += P V (V row-major: B frag = two b32 loads) ------------------
#pragma unroll
    for (int t = 0; t < 8; ++t) {
#pragma unroll
      for (int kc = 0; kc < 4; ++kc) {
        int r0 = kc * 4 + koff;
        v2f b;
        b.x = sV[r0 * STRIDEK + t * 16 + mrow];
        b.y = sV[(r0 + 1) * STRIDEK + t * 16 + mrow];
        acc[t] = __builtin_amdgcn_wmma_f32_16x16x4_f32(false, pf[kc], false, b,
                                                       (short)0, acc[t], false, false);
      }
    }
    buf ^= 1;
  }

  // --- dump per-wave partials (macc overlaps this wave's dead staging) ----
  float* macc = &smem[w * WSTAGE];   // [16][128]
#pragma unroll
  for (int t = 0; t < 8; ++t)
#pragma unroll
    for (int i = 0; i < 8; ++i) {
      int rl = (lane < 16) ? i : i + 8;
      macc[rl * 128 + t * 16 + mrow] = acc[t][i];
    }
  if (mrow == 0) {
#pragma unroll
    for (int i = 0; i < 8; ++i) {
      int rl = (lane < 16) ? i : i + 8;
      smem[OFF_ML + (w * 16 + rl) * 2 + 0] = m[i];
      smem[OFF_ML + (w * 16 + rl) * 2 + 1] = lsum[i];
    }
  }
  __syncthreads();

  // --- merge phase 1: per-row max / sum across waves ----------------------
  if (tid < NWAVE * 16) {
    int w2 = tid >> 4, r = tid & 15;
    float mstar = NEG_INF;
#pragma unroll
    for (int ww = 0; ww < NWAVE; ++ww)
      mstar = fmaxf(mstar, smem[OFF_ML + (ww * 16 + r) * 2 + 0]);
    smem[OFF_SC + w2 * 16 + r] =
        __expf(smem[OFF_ML + (w2 * 16 + r) * 2 + 0] - mstar);
    if (w2 == 0) {
      float lt = 0.f;
#pragma unroll
      for (int ww = 0; ww < NWAVE; ++ww)
        lt += smem[OFF_ML + (ww * 16 + r) * 2 + 1] *
              __expf(smem[OFF_ML + (ww * 16 + r) * 2 + 0] - mstar);
      smem[OFF_LS + r] = lt;
      smem[OFF_LE + r] = mstar + __logf(lt);
    }
  }
  __syncthreads();

  // --- merge phase 2: weighted combine + global writeback -----------------
  for (int it = 0; it < 16; ++it) {
    int o = tid + it * (32 * NWAVE);
    int r = o >> 7, d = o & 127;
    float num = 0.f;
#pragma unroll
    for (int ww = 0; ww < NWAVE; ++ww)
      num += smem[OFF_SC + ww * 16 + r] * smem[ww * WSTAGE + r * 128 + d];
    float val = num / smem[OFF_LS + r];
    int rg = rt * 16 + r;
    int g = rg >> 5, s = rg & 31;
    float* orow = out + ((s * HQ) + (kvh * GROUP + g)) * DHEAD;
    if (mode == 0) {
      orow[d] = val;
    } else {
      float lse_s = smem[OFF_LE + r];
      float lp = lse_p[kvh * 128 + rg];
      float mx = fmaxf(lp, lse_s);
      float lse = mx + __logf(__expf(lp - mx) + __expf(lse_s - mx));
      float wp = __expf(lp - lse);
      float wsx = __expf(lse_s - lse);
      orow[d] = wp * orow[d] + wsx * val;
    }
  }
  if (mode == 0 && tid < 16)
    lse_p[kvh * 128 + rt * 16 + tid] = smem[OFF_LE + tid];
}

// ---------------------------------------------------------------------------
extern "C" void kernel_launch(void* const* d_in, const int* in_sizes, int n_in,
                              void* d_out, int out_size, void* d_ws, size_t ws_size,
                              hipStream_t stream) {
  (void)in_sizes; (void)n_in; (void)out_size; (void)ws_size;
  const float* xq      = (const float*)d_in[0];
  const float* xk      = (const float*)d_in[1];
  const float* xv      = (const float*)d_in[2];
  const float* dram_k  = (const float*)d_in[3];
  const float* dram_v  = (const float*)d_in[4];
  const float* local_k = (const float*)d_in[5];
  const float* local_v = (const float*)d_in[6];
  float* out = (float*)d_out;

  float* ws     = (float*)d_ws;
  float* qsum   = ws;                    // 1024 f32
  float* scores = ws + 1024;             // 1024 f32
  int*   idx    = (int*)(ws + 2048);     //   64 i32
  float* lse    = ws + 2048 + 64;        // 1024 f32

  qsum_kernel <<<1,      256, 0, stream>>>(xq, qsum);
  score_kernel<<<NPAGES, 256, 0, stream>>>(dram_k, qsum, scores);
  topk_kernel <<<1,      256, 0, stream>>>(scores, idx);
  attn_kernel<<<64, 32 * NWAVE, 0, stream>>>(xq, dram_k, dram_v, nullptr, nullptr,
                                             idx, out, lse, 0);
  attn_kernel<<<64, 32 * NWAVE, 0, stream>>>(xq, local_k, local_v, xk, xv,
                                             idx, out, lse, 1);
}